// SABlock_28381143892091
// MI455X (gfx1250) — compile-verified
//
#include <hip/hip_runtime.h>
#include <hip/hip_bf16.h>

typedef float v2f __attribute__((ext_vector_type(2)));
typedef float v8f __attribute__((ext_vector_type(8)));

#define B_    8
#define N_    216
#define H_    12
#define HID_  768
#define D_    64
#define NPAD  224
#define NT    14            // 224/16 column tiles
#define BH    (B_*H_)       // 96
#define SCALE_ 0.125f       // 64^-0.5

// workspace layout (float offsets)
#define WS_QKSZ (BH*NPAD*D_)                 // 1,376,256 per tensor
#define WS_Q    0
#define WS_K    (WS_QKSZ)
#define WS_V    (2*WS_QKSZ)                  // stored TRANSPOSED: vt[bh][d][y]
#define WS_OMID (3*WS_QKSZ)                  // (B,N,HID) = 1,327,104
#define WS_SQK  (WS_OMID + B_*N_*HID_)       // (BH,N,4)  = 82,944
#define WS_KLD  (WS_SQK + BH*N_*4)           // BH*NT     = 1,344

__device__ __forceinline__ v8f wmma4(v2f a, v2f b, v8f c) {
  // D = A(16x4,f32) x B(4x16,f32) + C(16x16,f32)
  return __builtin_amdgcn_wmma_f32_16x16x4_f32(
      /*neg_a=*/false, a, /*neg_b=*/false, b,
      /*c_mod=*/(short)0, c, /*reuse_a=*/false, /*reuse_b=*/false);
}

// ---------------------------------------------------------------- pad rows
// zero rows 216..223 of q,k and columns 216..223 of vt
__global__ void pad_zero_kernel(float* ws) {
  int idx = blockIdx.x * 256 + threadIdx.x;
  const int total_qk = 2 * BH * 8 * D_;       // 98304
  const int total_v  = BH * D_ * 8;           // 49152
  if (idx < total_qk) {
    int which = idx / (BH * 8 * D_);
    int rem   = idx % (BH * 8 * D_);
    int bh    = rem / (8 * D_);
    int rd    = rem % (8 * D_);
    int r     = N_ + rd / D_;
    int d     = rd % D_;
    ws[which * WS_QKSZ + (bh * NPAD + r) * D_ + d] = 0.0f;
  } else if (idx < total_qk + total_v) {
    int i2 = idx - total_qk;
    int bh = i2 / (D_ * 8);
    int rem = i2 % (D_ * 8);
    int d  = rem / 8;
    int y  = N_ + rem % 8;
    ws[WS_V + (bh * D_ + d) * NPAD + y] = 0.0f;
  }
}

// ---------------------------------------------------------------- QKV GEMM
// C[m, c] = sum_k x[m,k] * w_qkv[c,k];  one wave -> 32x64 tile
// two-stage pipe with even/odd accumulators (one WMMA site per acc per iter)
__global__ __launch_bounds__(32) void qkv_gemm_kernel(
    const float* __restrict__ x, const float* __restrict__ w,
    float* __restrict__ ws) {
  const int mt   = blockIdx.x / 36;      // 54 tiles of 32 rows
  const int ntb  = blockIdx.x % 36;
  const int lane = threadIdx.x;
  const int lm   = lane & 15;
  const int half = lane >> 4;
  const int kl   = half * 2;

  const float* arow0 = x + (mt * 32 + lm) * HID_ + kl;
  const float* arow1 = arow0 + 16 * HID_;
  const float* brow  = w + (ntb * 64 + lm) * HID_ + kl;

  v8f zero = {0,0,0,0,0,0,0,0};
  v8f accE[2][4], accO[2][4];
#pragma unroll
  for (int r = 0; r < 2; ++r)
#pragma unroll
    for (int t = 0; t < 4; ++t) { accE[r][t] = zero; accO[r][t] = zero; }

  v2f a0S0, a1S0, a0S1, a1S1, bS0[4], bS1[4];
  a0S0 = *(const v2f*)(arow0);
  a1S0 = *(const v2f*)(arow1);
#pragma unroll
  for (int t = 0; t < 4; ++t) bS0[t] = *(const v2f*)(brow + t * 16 * HID_);

#pragma unroll 1
  for (int k0 = 0; k0 < HID_ - 8; k0 += 8) {
    a0S1 = *(const v2f*)(arow0 + k0 + 4);
    a1S1 = *(const v2f*)(arow1 + k0 + 4);
#pragma unroll
    for (int t = 0; t < 4; ++t) bS1[t] = *(const v2f*)(brow + t * 16 * HID_ + k0 + 4);
#pragma unroll
    for (int t = 0; t < 4; ++t) {
      accE[0][t] = wmma4(a0S0, bS0[t], accE[0][t]);
      accE[1][t] = wmma4(a1S0, bS0[t], accE[1][t]);
    }
    a0S0 = *(const v2f*)(arow0 + k0 + 8);
    a1S0 = *(const v2f*)(arow1 + k0 + 8);
#pragma unroll
    for (int t = 0; t < 4; ++t) bS0[t] = *(const v2f*)(brow + t * 16 * HID_ + k0 + 8);
#pragma unroll
    for (int t = 0; t < 4; ++t) {
      accO[0][t] = wmma4(a0S1, bS1[t], accO[0][t]);
      accO[1][t] = wmma4(a1S1, bS1[t], accO[1][t]);
    }
  }
  // epilogue: k = 760 in S0, load 764 into S1
  a0S1 = *(const v2f*)(arow0 + HID_ - 4);
  a1S1 = *(const v2f*)(arow1 + HID_ - 4);
#pragma unroll
  for (int t = 0; t < 4; ++t) bS1[t] = *(const v2f*)(brow + t * 16 * HID_ + HID_ - 4);
#pragma unroll
  for (int t = 0; t < 4; ++t) {
    accE[0][t] = wmma4(a0S0, bS0[t], accE[0][t]);
    accE[1][t] = wmma4(a1S0, bS0[t], accE[1][t]);
  }
#pragma unroll
  for (int t = 0; t < 4; ++t) {
    accO[0][t] = wmma4(a0S1, bS1[t], accO[0][t]);
    accO[1][t] = wmma4(a1S1, bS1[t], accO[1][t]);
  }

#pragma unroll
  for (int r = 0; r < 2; ++r) {
#pragma unroll
    for (int t = 0; t < 4; ++t) {
      v8f acc = accE[r][t] + accO[r][t];
      const int c     = ntb * 64 + t * 16 + lm;
      const int three = c / 768;
      const int rem   = c % 768;
      const int h     = rem >> 6;
      const int d     = rem & 63;
#pragma unroll
      for (int j = 0; j < 8; ++j) {
        const int m = mt * 32 + r * 16 + j + 8 * half;
        const int b = m / N_;
        const int n = m % N_;
        float val = acc[j];
        if (three == 2)   // v stored transposed: vt[bh][d][y]
          ws[WS_V + ((b * H_ + h) * D_ + d) * NPAD + n] = val;
        else
          ws[three * WS_QKSZ + ((b * H_ + h) * NPAD + n) * D_ + d] = val;
      }
    }
  }
}

// ---------------------------------------------------------------- sq / sk
__global__ void sqk_kernel(float* ws, const float* __restrict__ fc_w) {
  int idx = blockIdx.x * 256 + threadIdx.x;
  if (idx >= BH * N_) return;
  const int bh = idx / N_;
  const int n  = idx % N_;
  const float* q = ws + WS_Q + (bh * NPAD + n) * D_;
  const float* k = ws + WS_K + (bh * NPAD + n) * D_;
  float s0 = 0.f, s1 = 0.f, s2 = 0.f, s3 = 0.f;
  for (int d = 0; d < D_; ++d) {
    float qd = q[d], kd = k[d];
    s0 += qd * fc_w[d];             // wq row0
    s1 += qd * fc_w[128 + d];       // wq row1
    s2 += kd * fc_w[64 + d];        // wk row0
    s3 += kd * fc_w[192 + d];       // wk row1
  }
  float* o = ws + WS_SQK + idx * 4;
  o[0] = s0; o[1] = s1; o[2] = s2; o[3] = s3;
}

// ---------------------------------------------------------------- attention
// one wave per (b,h, 16-row tile): det -> logits -> softmax -> att@v
__global__ __launch_bounds__(32) void attn_kernel(
    const float* __restrict__ eps, const float* __restrict__ fc_b,
    float* ws) {
  __shared__ float att_lds[16 * NPAD];
  __shared__ float sk0[NPAD], sk1[NPAD], sq0[16], sq1[16];

  const int bh   = blockIdx.x / NT;
  const int tile = blockIdx.x % NT;
  const int lane = threadIdx.x;
  const int lm   = lane & 15;
  const int half = lane >> 4;
  const int kl   = half * 2;
  const int row0 = tile * 16;

  const float* qb  = ws + WS_Q + (bh * NPAD) * D_;
  const float* kb  = ws + WS_K + (bh * NPAD) * D_;
  const float* vtb = ws + WS_V + bh * (D_ * NPAD);
  const float* sqk = ws + WS_SQK + bh * N_ * 4;

  for (int y = lane; y < NPAD; y += 32) {
    if (y < N_) { sk0[y] = sqk[y * 4 + 2]; sk1[y] = sqk[y * 4 + 3]; }
    else        { sk0[y] = 0.f;            sk1[y] = 0.f; }
  }
  if (lane < 16) {
    int r = row0 + lane;
    if (r < N_) { sq0[lane] = sqk[r * 4]; sq1[lane] = sqk[r * 4 + 1]; }
    else        { sq0[lane] = 0.f;        sq1[lane] = 0.f; }
  }
  __syncthreads();

  v8f zero = {0,0,0,0,0,0,0,0};
  v8f det[NT];

  // det = q_tile(16xD) @ k^T(Dx224): two passes of 7 column tiles,
  // each pass pipelined with even/odd accumulators
  const float* qrow = qb + (row0 + lm) * D_ + kl;
#pragma unroll
  for (int g = 0; g < 2; ++g) {
    const float* krow = kb + (g * 7 * 16 + lm) * D_ + kl;
    v8f accE[7], accO[7];
#pragma unroll
    for (int t = 0; t < 7; ++t) { accE[t] = zero; accO[t] = zero; }
    v2f aS0, aS1, bS0[7], bS1[7];
    aS0 = *(const v2f*)(qrow);
#pragma unroll
    for (int t = 0; t < 7; ++t) bS0[t] = *(const v2f*)(krow + t * 16 * D_);
#pragma unroll 1
    for (int d0 = 0; d0 < D_ - 8; d0 += 8) {
      aS1 = *(const v2f*)(qrow + d0 + 4);
#pragma unroll
      for (int t = 0; t < 7; ++t) bS1[t] = *(const v2f*)(krow + t * 16 * D_ + d0 + 4);
#pragma unroll
      for (int t = 0; t < 7; ++t) accE[t] = wmma4(aS0, bS0[t], accE[t]);
      aS0 = *(const v2f*)(qrow + d0 + 8);
#pragma unroll
      for (int t = 0; t < 7; ++t) bS0[t] = *(const v2f*)(krow + t * 16 * D_ + d0 + 8);
#pragma unroll
      for (int t = 0; t < 7; ++t) accO[t] = wmma4(aS1, bS1[t], accO[t]);
    }
    aS1 = *(const v2f*)(qrow + D_ - 4);
#pragma unroll
    for (int t = 0; t < 7; ++t) bS1[t] = *(const v2f*)(krow + t * 16 * D_ + D_ - 4);
#pragma unroll
    for (int t = 0; t < 7; ++t) accE[t] = wmma4(aS0, bS0[t], accE[t]);
#pragma unroll
    for (int t = 0; t < 7; ++t) accO[t] = wmma4(aS1, bS1[t], accO[t]);
#pragma unroll
    for (int t = 0; t < 7; ++t) det[g * 7 + t] = accE[t] + accO[t];
  }

  const float fb0 = fc_b[0], fb1 = fc_b[1];
  const float* epsb = eps + bh * N_ * N_;
  const float NEGINF = -__builtin_inff();
  float kacc = 0.0f;

  // logits + KLD, overwrite det accumulators with logits (C layout)
#pragma unroll
  for (int j = 0; j < 8; ++j) {
    const int m = j + 8 * half;
    const int r = row0 + m;
#pragma unroll
    for (int t = 0; t < NT; ++t) {
      const int y = t * 16 + lm;
      float logit;
      if (r < N_ && y < N_) {
        float dd = det[t][j] * SCALE_;
        float mu = sq0[m] + sk0[y] + fb0;
        float lv = sq1[m] + sk1[y] + fb1;
        float e  = epsb[r * N_ + y];
        logit = e * __expf(0.5f * lv) + mu;
        float diff = mu - dd;
        kacc += -0.5f * (1.0f + lv - (diff * diff + __expf(lv)));
      } else {
        logit = NEGINF;
      }
      det[t][j] = logit;
    }
  }

  // row softmax: rows of one m live in 14 regs x 16 lanes of one half-wave
#pragma unroll
  for (int j = 0; j < 8; ++j) {
    float mx = NEGINF;
#pragma unroll
    for (int t = 0; t < NT; ++t) mx = fmaxf(mx, det[t][j]);
#pragma unroll
    for (int mk = 1; mk < 16; mk <<= 1) mx = fmaxf(mx, __shfl_xor(mx, mk, 32));
    float s = 0.0f;
#pragma unroll
    for (int t = 0; t < NT; ++t) {
      float l  = det[t][j];
      float ex = (l == NEGINF) ? 0.0f : __expf(l - mx);
      det[t][j] = ex;
      s += ex;
    }
#pragma unroll
    for (int mk = 1; mk < 16; mk <<= 1) s += __shfl_xor(s, mk, 32);
    float inv = (s > 0.0f) ? 1.0f / s : 0.0f;
    const int m = j + 8 * half;
#pragma unroll
    for (int t = 0; t < NT; ++t)
      att_lds[m * NPAD + t * 16 + lm] = det[t][j] * inv;
  }

  // deterministic KLD partial per block
#pragma unroll
  for (int mk = 1; mk < 32; mk <<= 1) kacc += __shfl_xor(kacc, mk, 32);
  if (lane == 0) ws[WS_KLD + blockIdx.x] = kacc;
  __syncthreads();

  // o = att(16x224) @ v(224x64); vt gives k-contiguous B frags;
  // pipelined with even/odd accumulators
  v8f oacc[4];
  {
    const float* arow  = att_lds + lm * NPAD + kl;
    const float* vtrow = vtb + lm * NPAD + kl;
    v8f accE[4], accO[4];
#pragma unroll
    for (int t = 0; t < 4; ++t) { accE[t] = zero; accO[t] = zero; }
    v2f aS0, aS1, bS0[4], bS1[4];
    aS0 = *(const v2f*)(arow);
#pragma unroll
    for (int t = 0; t < 4; ++t) bS0[t] = *(const v2f*)(vtrow + t * 16 * NPAD);
#pragma unroll 1
    for (int y0 = 0; y0 < NPAD - 8; y0 += 8) {
      aS1 = *(const v2f*)(arow + y0 + 4);
#pragma unroll
      for (int t = 0; t < 4; ++t) bS1[t] = *(const v2f*)(vtrow + t * 16 * NPAD + y0 + 4);
#pragma unroll
      for (int t = 0; t < 4; ++t) accE[t] = wmma4(aS0, bS0[t], accE[t]);
      aS0 = *(const v2f*)(arow + y0 + 8);
#pragma unroll
      for (int t = 0; t < 4; ++t) bS0[t] = *(const v2f*)(vtrow + t * 16 * NPAD + y0 + 8);
#pragma unroll
      for (int t = 0; t < 4; ++t) accO[t] = wmma4(aS1, bS1[t], accO[t]);
    }
    aS1 = *(const v2f*)(arow + NPAD - 4);
#pragma unroll
    for (int t = 0; t < 4; ++t) bS1[t] = *(const v2f*)(vtrow + t * 16 * NPAD + NPAD - 4);
#pragma unroll
    for (int t = 0; t < 4; ++t) accE[t] = wmma4(aS0, bS0[t], accE[t]);
#pragma unroll
    for (int t = 0; t < 4; ++t) accO[t] = wmma4(aS1, bS1[t], accO[t]);
#pragma unroll
    for (int t = 0; t < 4; ++t) oacc[t] = accE[t] + accO[t];
  }

  const int bb = bh / H_, hh = bh % H_;
  float* ob = ws + WS_OMID;
#pragma unroll
  for (int j = 0; j < 8; ++j) {
    const int m = j + 8 * half;
    const int r = row0 + m;
    if (r < N_) {
#pragma unroll
      for (int t = 0; t < 4; ++t)
        ob[(bb * N_ + r) * HID_ + hh * D_ + t * 16 + lm] = oacc[t][j];
    }
  }
}

// ---------------------------------------------------------------- nll
__global__ void nll_kernel(const float* __restrict__ kld, float* out_nll) {
  int b = threadIdx.x;
  if (b < B_) {
    float s = 0.0f;
    for (int i = 0; i < H_ * NT; ++i) s += kld[b * H_ * NT + i];
    out_nll[b] = s * (1.0f / (float)(H_ * N_ * N_));
  }
}

// ---------------------------------------------------------------- out proj
// out[m,n] = sum_k omid[m,k] * out_w[n,k] + out_b[n]; 32x64 tile per wave
__global__ __launch_bounds__(32) void proj_gemm_kernel(
    const float* __restrict__ omid, const float* __restrict__ w,
    const float* __restrict__ bias, float* __restrict__ out) {
  const int mt   = blockIdx.x / 12;      // 54 tiles of 32 rows
  const int ntb  = blockIdx.x % 12;
  const int lane = threadIdx.x;
  const int lm   = lane & 15;
  const int half = lane >> 4;
  const int kl   = half * 2;

  const float* arow0 = omid + (mt * 32 + lm) * HID_ + kl;
  const float* arow1 = arow0 + 16 * HID_;
  const float* brow  = w + (ntb * 64 + lm) * HID_ + kl;

  v8f zero = {0,0,0,0,0,0,0,0};
  v8f accE[2][4], accO[2][4];
#pragma unroll
  for (int r = 0; r < 2; ++r)
#pragma unroll
    for (int t = 0; t < 4; ++t) { accE[r][t] = zero; accO[r][t] = zero; }

  v2f a0S0, a1S0, a0S1, a1S1, bS0[4], bS1[4];
  a0S0 = *(const v2f*)(arow0);
  a1S0 = *(const v2f*)(arow1);
#pragma unroll
  for (int t = 0; t < 4; ++t) bS0[t] = *(const v2f*)(brow + t * 16 * HID_);

#pragma unroll 1
  for (int k0 = 0; k0 < HID_ - 8; k0 += 8) {
    a0S1 = *(const v2f*)(arow0 + k0 + 4);
    a1S1 = *(const v2f*)(arow1 + k0 + 4);
#pragma unroll
    for (int t = 0; t < 4; ++t) bS1[t] = *(const v2f*)(brow + t * 16 * HID_ + k0 + 4);
#pragma unroll
    for (int t = 0; t < 4; ++t) {
      accE[0][t] = wmma4(a0S0, bS0[t], accE[0][t]);
      accE[1][t] = wmma4(a1S0, bS0[t], accE[1][t]);
    }
    a0S0 = *(const v2f*)(arow0 + k0 + 8);
    a1S0 = *(const v2f*)(arow1 + k0 + 8);
#pragma unroll
    for (int t = 0; t < 4; ++t) bS0[t] = *(const v2f*)(brow + t * 16 * HID_ + k0 + 8);
#pragma unroll
    for (int t = 0; t < 4; ++t) {
      accO[0][t] = wmma4(a0S1, bS1[t], accO[0][t]);
      accO[1][t] = wmma4(a1S1, bS1[t], accO[1][t]);
    }
  }
  a0S1 = *(const v2f*)(arow0 + HID_ - 4);
  a1S1 = *(const v2f*)(arow1 + HID_ - 4);
#pragma unroll
  for (int t = 0; t < 4; ++t) bS1[t] = *(const v2f*)(brow + t * 16 * HID_ + HID_ - 4);
#pragma unroll
  for (int t = 0; t < 4; ++t) {
    accE[0][t] = wmma4(a0S0, bS0[t], accE[0][t]);
    accE[1][t] = wmma4(a1S0, bS0[t], accE[1][t]);
  }
#pragma unroll
  for (int t = 0; t < 4; ++t) {
    accO[0][t] = wmma4(a0S1, bS1[t], accO[0][t]);
    accO[1][t] = wmma4(a1S1, bS1[t], accO[1][t]);
  }

#pragma unroll
  for (int r = 0; r < 2; ++r) {
#pragma unroll
    for (int t = 0; t < 4; ++t) {
      v8f acc = accE[r][t] + accO[r][t];
      const int n  = ntb * 64 + t * 16 + lm;
      const float bn = bias[n];
#pragma unroll
      for (int j = 0; j < 8; ++j) {
        const int m = mt * 32 + r * 16 + j + 8 * half;
        out[m * HID_ + n] = acc[j] + bn;
      }
    }
  }
}

extern "C" void kernel_launch(void* const* d_in, const int* in_sizes, int n_in,
                              void* d_out, int out_size, void* d_ws, size_t ws_size,
                              hipStream_t stream) {
  const float* x     = (const float*)d_in[0];
  const float* w_qkv = (const float*)d_in[1];
  const float* fc_w  = (const float*)d_in[2];
  const float* fc_b  = (const float*)d_in[3];
  const float* out_w = (const float*)d_in[4];
  const float* out_b = (const float*)d_in[5];
  const float* eps   = (const float*)d_in[6];
  float* ws  = (float*)d_ws;
  float* out = (float*)d_out;

  pad_zero_kernel<<<(2 * BH * 8 * D_ + BH * D_ * 8 + 255) / 256, 256, 0, stream>>>(ws);
  qkv_gemm_kernel<<<54 * 36, 32, 0, stream>>>(x, w_qkv, ws);
  sqk_kernel<<<(BH * N_ + 255) / 256, 256, 0, stream>>>(ws, fc_w);
  attn_kernel<<<BH * NT, 32, 0, stream>>>(eps, fc_b, ws);
  nll_kernel<<<1, 32, 0, stream>>>(ws + WS_KLD, out + B_ * N_ * HID_);
  proj_gemm_kernel<<<54 * 12, 32, 0, stream>>>(ws + WS_OMID, out_w, out_b, out);
}